// Gcn_86612310492049
// MI455X (gfx1250) — compile-verified
//
#include <hip/hip_runtime.h>
#include <math.h>

typedef __attribute__((ext_vector_type(2))) float v2f;
typedef __attribute__((ext_vector_type(8))) float v8f;

#define F_IN  256
#define F_HID 128
#define F_OUT 64

// ---------------------------------------------------------------------------
// Degree / normalization
// ---------------------------------------------------------------------------
__global__ void k_deg_init(float* __restrict__ deg, int n) {
    int t = blockIdx.x * blockDim.x + threadIdx.x;
    if (t < n) deg[t] = 1.0f;                 // self-loop contribution
}

__global__ void k_deg_count(float* __restrict__ deg, const int* __restrict__ col, int e) {
    int t = blockIdx.x * blockDim.x + threadIdx.x;
    if (t < e) atomicAdd(&deg[col[t]], 1.0f);
}

__global__ void k_rsqrt(const float* __restrict__ deg, float* __restrict__ dis, int n) {
    int t = blockIdx.x * blockDim.x + threadIdx.x;
    if (t < n) dis[t] = rsqrtf(deg[t]);       // deg >= 1 always (self loops)
}

// ---------------------------------------------------------------------------
// Dense GEMM: C[n_rows x N] = A[n_rows x K] @ W[K x N], fp32 via
// v_wmma_f32_16x16x4_f32. One wave handles a 16-row tile and the full N
// (N/16 accumulators). Requires n_rows % 16 == 0 (100000 = 6250*16).
// A-fragment (16x4 f32): lanes 0-15 hold M=0..15; VGPR0/1 = K pair; lane
// halves split K {0,1} vs {2,3}. B-fragment (4x16): N striped on lanes,
// same K split. C/D: VGPR v -> row v (+8 for upper lane half), N on lanes.
// ---------------------------------------------------------------------------
template <int K, int N>
__global__ void k_gemm_wmma(const float* __restrict__ A, const float* __restrict__ W,
                            float* __restrict__ C, int n_rows) {
    constexpr int NT = N / 16;
    const int wave = (int)((blockIdx.x * blockDim.x + threadIdx.x) >> 5);
    const int lane = (int)(threadIdx.x & 31);
    const int tile = wave;                      // 16-row node tile
    if (tile * 16 >= n_rows) return;            // wave-uniform: EXEC stays all-1s

    const int m     = lane & 15;                // row within tile (A/B lane id)
    const int khalf = lane >> 4;                // 0: K{0,1}, 1: K{2,3}

    v8f acc[NT] = {};                           // zero accumulators

    const float* arow = A + (size_t)(tile * 16 + m) * K;

    for (int k0 = 0; k0 < K; k0 += 4) {
        const int kk = k0 + khalf * 2;
        v2f a = *(const v2f*)(arow + kk);       // A[m][kk], A[m][kk+1]
#pragma unroll
        for (int nt = 0; nt < NT; ++nt) {
            const float* bp = W + (size_t)kk * N + nt * 16 + m;
            v2f b;
            b.x = bp[0];                        // W[kk][col]
            b.y = bp[N];                        // W[kk+1][col]
            acc[nt] = __builtin_amdgcn_wmma_f32_16x16x4_f32(
                false, a, false, b, (short)0, acc[nt], false, false);
        }
    }

    const int mbase = tile * 16 + khalf * 8;
#pragma unroll
    for (int nt = 0; nt < NT; ++nt) {
#pragma unroll
        for (int v = 0; v < 8; ++v) {
            C[(size_t)(mbase + v) * N + nt * 16 + m] = acc[nt][v];
        }
    }
}

// ---------------------------------------------------------------------------
// out[i][:] = b[:] + dis[i]^2 * h[i][:]   (bias + self-loop message; also the
// initializer so no separate memset of the accumulation buffer is needed)
// ---------------------------------------------------------------------------
template <int F>
__global__ void k_init_agg(const float* __restrict__ h, const float* __restrict__ dis,
                           const float* __restrict__ bias, float* __restrict__ out, int n) {
    constexpr int CPN = F / 4;                  // float4 chunks per node
    int t = blockIdx.x * blockDim.x + threadIdx.x;
    int node = t / CPN;
    if (node >= n) return;
    int j = (t % CPN) * 4;
    float s = dis[node];
    s = s * s;
    float4 hv = *(const float4*)(h + (size_t)node * F + j);
    float4 bv = *(const float4*)(bias + j);
    float4 o;
    o.x = bv.x + s * hv.x;
    o.y = bv.y + s * hv.y;
    o.z = bv.z + s * hv.z;
    o.w = bv.w + s * hv.w;
    *(float4*)(out + (size_t)node * F + j) = o;
}

// ---------------------------------------------------------------------------
// Edge scatter: one wave32 per edge, lane l moves features [4l, 4l+4).
// out[c][:] += dis[r]*dis[c] * h[r][:]
// ---------------------------------------------------------------------------
template <int F>
__global__ void k_scatter(const float* __restrict__ h, const int* __restrict__ rows,
                          const int* __restrict__ cols, const float* __restrict__ dis,
                          float* __restrict__ out, int e) {
    static_assert(F == 128, "one float4 per lane");
    const int wid  = (int)((blockIdx.x * blockDim.x + threadIdx.x) >> 5);
    const int lane = (int)(threadIdx.x & 31);
    if (wid >= e) return;
    const int r = rows[wid];
    const int c = cols[wid];
    const float nrm = dis[r] * dis[c];
    const float4 v = *(const float4*)(h + (size_t)r * F + lane * 4);
    float* o = out + (size_t)c * F + lane * 4;
    atomicAdd(o + 0, nrm * v.x);
    atomicAdd(o + 1, nrm * v.y);
    atomicAdd(o + 2, nrm * v.z);
    atomicAdd(o + 3, nrm * v.w);
}

// ---------------------------------------------------------------------------
// In-place ReLU over n4 float4's
// ---------------------------------------------------------------------------
__global__ void k_relu4(float* __restrict__ x, int n4) {
    int t = blockIdx.x * blockDim.x + threadIdx.x;
    if (t >= n4) return;
    float4 v = *(const float4*)(x + (size_t)t * 4);
    v.x = fmaxf(v.x, 0.0f);
    v.y = fmaxf(v.y, 0.0f);
    v.z = fmaxf(v.z, 0.0f);
    v.w = fmaxf(v.w, 0.0f);
    *(float4*)(x + (size_t)t * 4) = v;
}

// ---------------------------------------------------------------------------
// In-place log_softmax over 64 logits per node (+ bias). One wave per node,
// 2 features per lane, shfl_xor reductions across the wave32.
// ---------------------------------------------------------------------------
__global__ void k_logsoftmax64(float* __restrict__ logits, const float* __restrict__ b3, int n) {
    const int wid  = (int)((blockIdx.x * blockDim.x + threadIdx.x) >> 5);
    const int lane = (int)(threadIdx.x & 31);
    if (wid >= n) return;
    float* p = logits + (size_t)wid * 64 + lane * 2;
    float2 v = *(const float2*)p;
    float2 bv = *(const float2*)(b3 + lane * 2);
    v.x += bv.x;
    v.y += bv.y;
    float m = fmaxf(v.x, v.y);
#pragma unroll
    for (int o = 16; o > 0; o >>= 1) m = fmaxf(m, __shfl_xor(m, o, 32));
    float s = expf(v.x - m) + expf(v.y - m);
#pragma unroll
    for (int o = 16; o > 0; o >>= 1) s += __shfl_xor(s, o, 32);
    const float lse = m + logf(s);
    float2 r;
    r.x = v.x - lse;
    r.y = v.y - lse;
    *(float2*)p = r;
}

// ---------------------------------------------------------------------------
// Launch
// ---------------------------------------------------------------------------
static inline int cdiv(long long a, long long b) { return (int)((a + b - 1) / b); }

extern "C" void kernel_launch(void* const* d_in, const int* in_sizes, int n_in,
                              void* d_out, int out_size, void* d_ws, size_t ws_size,
                              hipStream_t stream) {
    const float* x  = (const float*)d_in[0];
    const int*   ei = (const int*)d_in[1];
    const float* W1 = (const float*)d_in[2];
    const float* b1 = (const float*)d_in[3];
    const float* W2 = (const float*)d_in[4];
    const float* b2 = (const float*)d_in[5];
    const float* W3 = (const float*)d_in[6];
    const float* b3 = (const float*)d_in[7];

    const int n = in_sizes[0] / F_IN;          // 100000
    const int e = in_sizes[1] / 2;             // 1600000
    const int* rows = ei;                      // edge_index[0]
    const int* cols = ei + e;                  // edge_index[1]

    float* ws   = (float*)d_ws;
    float* deg  = ws;                          // [n]
    float* dis  = ws + (1 << 17);              // [n]
    float* bufA = ws + (1 << 18);              // [n * 128]
    float* bufB = bufA + (size_t)n * F_HID;    // [n * 128]
    float* logits = (float*)d_out;             // [n * 64]

    const int B = 256;
    const int wavesPerBlock = B / 32;

    // degree + rsqrt
    k_deg_init<<<cdiv(n, B), B, 0, stream>>>(deg, n);
    k_deg_count<<<cdiv(e, B), B, 0, stream>>>(deg, cols, e);
    k_rsqrt<<<cdiv(n, B), B, 0, stream>>>(deg, dis, n);

    const int gemmBlocks = cdiv((long long)n / 16, wavesPerBlock);

    // ---- layer 1: h1 = relu(agg(x @ W1) + b1) ----
    k_gemm_wmma<F_IN, F_HID><<<gemmBlocks, B, 0, stream>>>(x, W1, bufA, n);
    k_init_agg<F_HID><<<cdiv((long long)n * (F_HID / 4), B), B, 0, stream>>>(bufA, dis, b1, bufB, n);
    k_scatter<F_HID><<<cdiv(e, wavesPerBlock), B, 0, stream>>>(bufA, rows, cols, dis, bufB, e);
    k_relu4<<<cdiv((long long)n * F_HID / 4, B), B, 0, stream>>>(bufB, n * F_HID / 4);

    // ---- layer 2: h2 = relu(agg(h1 @ W2) + b2) ----
    k_gemm_wmma<F_HID, F_HID><<<gemmBlocks, B, 0, stream>>>(bufB, W2, bufA, n);
    k_init_agg<F_HID><<<cdiv((long long)n * (F_HID / 4), B), B, 0, stream>>>(bufA, dis, b2, bufB, n);
    k_scatter<F_HID><<<cdiv(e, wavesPerBlock), B, 0, stream>>>(bufA, rows, cols, dis, bufB, e);
    k_relu4<<<cdiv((long long)n * F_HID / 4, B), B, 0, stream>>>(bufB, n * F_HID / 4);

    // ---- head: log_softmax(h2 @ W3 + b3) ----
    k_gemm_wmma<F_HID, F_OUT><<<gemmBlocks, B, 0, stream>>>(bufB, W3, logits, n);
    k_logsoftmax64<<<cdiv(n, wavesPerBlock), B, 0, stream>>>(logits, b3, n);
}